// CNOT_6992206758255
// MI455X (gfx1250) — compile-verified
//
#include <hip/hip_runtime.h>
#include <hip/hip_bf16.h>
#include <stdint.h>

// ---------------------------------------------------------------------------
// Workload: out[perm[k]*16 + c] = x[k*16 + c]   (row permutation, fp32,
// 4782969 rows x 16 cols). 0 FLOPs, ~631 MB of traffic -> pure HBM-bandwidth
// problem (~27 us floor at 23.3 TB/s). Optimal path on MI455X: coalesced
// B128 non-temporal streaming loads/stores, one 16B chunk per lane.
// ---------------------------------------------------------------------------

typedef __attribute__((ext_vector_type(4))) float v4f;

// Fast path: batch == 16 floats/row -> 4 x 16B chunks per row.
// Thread j moves chunk (k = j>>2, c = j&3). Reads fully coalesced; writes
// coalesced too because perm is a block shuffle (blocks of 3^12 rows).
__global__ __launch_bounds__(256) void perm_scatter_b128(
    const v4f* __restrict__ x, const int* __restrict__ perm,
    v4f* __restrict__ out, int nchunks) {
  int j = blockIdx.x * 256 + threadIdx.x;
  if (j >= nchunks) return;
  int k = j >> 2;                       // source row
  int c = j & 3;                        // 16B chunk within row
  v4f v = __builtin_nontemporal_load(&x[j]);      // NT: 306MB stream > L2
  int pk = perm[k];                     // temporal: 1 cacheline feeds 64 lanes
  int oj = (pk << 2) | c;
  __builtin_nontemporal_store(v, &out[oj]);       // NT: 306MB stream > L2
}

// Generic fallback for any batch width (scalar, still coalesced).
__global__ __launch_bounds__(256) void perm_scatter_scalar(
    const float* __restrict__ x, const int* __restrict__ perm,
    float* __restrict__ out, long long n, int batch) {
  long long j = (long long)blockIdx.x * 256 + threadIdx.x;
  if (j >= n) return;
  long long k = j / batch;
  int c = (int)(j - k * (long long)batch);
  float v = __builtin_nontemporal_load(&x[j]);
  long long oj = (long long)perm[k] * batch + c;
  __builtin_nontemporal_store(v, &out[oj]);
}

// ---------------------------------------------------------------------------
// Probe kernels (compiled, never launched): exercise the gfx1250-only async
// data movers so the histogram/disasm confirms syntax for future use.
// ---------------------------------------------------------------------------

// ASYNCcnt path: global -> LDS -> global without touching VGPR data.
__global__ void probe_async_lds_copy(const float* __restrict__ g,
                                     float* __restrict__ o) {
  __shared__ float lbuf[256];
  unsigned lds_off = threadIdx.x * 4u;
  const float* gp = g + threadIdx.x;
  float* op = o + threadIdx.x;
  asm volatile("global_load_async_to_lds_b32 %0, %1, off"
               :: "v"(lds_off), "v"(gp) : "memory");
  asm volatile("s_wait_asynccnt 0" ::: "memory");
  asm volatile("global_store_async_from_lds_b32 %0, %1, off"
               :: "v"(op), "v"(lds_off) : "memory");
  asm volatile("s_wait_asynccnt 0" ::: "memory");
  if (g == nullptr) o[0] = lbuf[0];   // keep lbuf allocated
}

// TENSORcnt path: TDM descriptor-driven DMA + cluster machinery.
// This toolchain (clang-23 / therock-10.0) uses the 6-arg builtin:
//   (uint32x4 g0, int32x8 g1, int32x4 g2, int32x4 g3, int32x8 g4, i32 cpol)
#if __has_builtin(__builtin_amdgcn_tensor_load_to_lds)
typedef __attribute__((ext_vector_type(4))) unsigned int u32x4;
typedef __attribute__((ext_vector_type(8))) int i32x8;
typedef __attribute__((ext_vector_type(4))) int i32x4;
__global__ void probe_tdm(int* __restrict__ sink) {
  u32x4 g0 = {0u, 0u, 0u, 0u};
  i32x8 g1 = {0, 0, 0, 0, 0, 0, 0, 0};
  i32x4 g2 = {0, 0, 0, 0};
  i32x4 g3 = {0, 0, 0, 0};
  i32x8 g4 = {0, 0, 0, 0, 0, 0, 0, 0};
  __builtin_amdgcn_tensor_load_to_lds(g0, g1, g2, g3, g4, 0);
  __builtin_amdgcn_s_wait_tensorcnt(0);
  if (sink) *sink = __builtin_amdgcn_cluster_id_x();
}
#endif

// ---------------------------------------------------------------------------

extern "C" void kernel_launch(void* const* d_in, const int* in_sizes, int n_in,
                              void* d_out, int out_size, void* d_ws, size_t ws_size,
                              hipStream_t stream) {
  const float* x   = (const float*)d_in[0];   // (DSTATE, BATCH) fp32
  const int* perm  = (const int*)d_in[1];     // (DSTATE,) int32
  float* out       = (float*)d_out;

  long long rows  = (long long)in_sizes[1];
  long long total = (long long)in_sizes[0];
  int batch = (int)(total / rows);            // 16 for the reference

  if (batch == 16) {
    int nchunks = (int)(total >> 2);          // 19,131,876 x 16B chunks
    int blocks = (nchunks + 255) / 256;
    perm_scatter_b128<<<blocks, 256, 0, stream>>>(
        (const v4f*)x, perm, (v4f*)out, nchunks);
  } else {
    long long blocks = (total + 255) / 256;
    perm_scatter_scalar<<<(int)blocks, 256, 0, stream>>>(
        x, perm, out, total, batch);
  }
}